// PGNNLayer_8924942041314
// MI455X (gfx1250) — compile-verified
//
#include <hip/hip_runtime.h>

typedef __attribute__((ext_vector_type(2))) float v2f;
typedef __attribute__((ext_vector_type(8))) float v8f;

#define NN    10000
#define BB    2
#define FF    64
#define NASZ  64
#define OUTC  64
#define KDIM  128          // 2*F
#define NPW   32           // n-values per workgroup
#define ROWS  64           // NPW * BB rows per workgroup
#define LSTR  132          // padded LDS row stride (floats), even -> 8B-aligned rows
#define THREADS 256        // 8 waves

__launch_bounds__(THREADS)
__global__ void pgnn_fused_kernel(const float* __restrict__ x,
                                  const float* __restrict__ dm,
                                  const int*   __restrict__ idx,
                                  const float* __restrict__ W,
                                  const float* __restrict__ bias,
                                  float* __restrict__ out)
{
    __shared__ float w_lds[OUTC * LSTR];   // W_hidden, row-major [o][k], padded
    __shared__ float h_lds[ROWS * LSTR];   // features [row][k]: k<64 = abar, k>=64 = x-row
    __shared__ float bias_lds[OUTC];
    __shared__ int   idx_lds[NPW * NASZ];
    __shared__ float dm_lds [NPW * NASZ];

    const int tid = threadIdx.x;
    const int n0  = blockIdx.x * NPW;

    // ---- preload weights + bias into LDS (coalesced) ----
    for (int e = tid; e < OUTC * KDIM; e += THREADS) {
        int o = e >> 7, k = e & 127;
        w_lds[o * LSTR + k] = W[e];
    }
    if (tid < OUTC) bias_lds[tid] = bias[tid];

    // ---- preload anchor indices / weights for this WG's 32 n's ----
    for (int e = tid; e < NPW * NASZ; e += THREADS) {
        int nl = e >> 6;
        int n  = n0 + nl;
        if (n < NN) {
            idx_lds[e] = idx[n * NASZ + (e & 63)];
            dm_lds[e]  = dm [n * NASZ + (e & 63)];
        } else {
            idx_lds[e] = 0;
            dm_lds[e]  = 0.0f;   // zero weight -> abar = 0 for pad rows
        }
    }
    __syncthreads();

    // ---- Stage A: weighted gather-average (x is L2-resident: 5.1 MB in 192 MB L2) ----
    // 256 threads = two 128-thread halves; half h handles nl in [16h, 16h+16).
    // Within a half, thread t = b*64 + f owns one (b,f); x[j,b,f] = x[j*128 + t].
    {
        const int half = tid >> 7;         // 0 or 1
        const int t    = tid & 127;
        const int b    = t >> 6;
        const int f    = t & 63;
        const int nlo  = half * (NPW / 2);
        for (int nl = nlo; nl < nlo + NPW / 2; ++nl) {
            const int*   ip = &idx_lds[nl * NASZ];
            const float* dp = &dm_lds [nl * NASZ];
            float acc = 0.0f;
            #pragma unroll 8
            for (int a = 0; a < NASZ; ++a) {
                acc = fmaf(dp[a], x[(size_t)ip[a] * KDIM + t], acc);
            }
            const int n   = n0 + nl;
            const int row = nl * 2 + b;
            h_lds[row * LSTR + f]      = acc * (1.0f / NASZ);
            h_lds[row * LSTR + FF + f] = (n < NN) ? x[(size_t)n * KDIM + t] : 0.0f;
        }
    }
    __syncthreads();

    // ---- Stage B: 8 waves; wave w -> row-tile (w>>1), col half (w&1) = 2 D tiles ----
    // D(16x16,f32) = A(16x4,f32) x B(4x16,f32) + C  via v_wmma_f32_16x16x4_f32
    const int wave  = tid >> 5;
    const int lane  = tid & 31;
    const int l15   = lane & 15;
    const int khalf = (lane >> 4) * 2;     // lanes 0-15: K pair {0,1}; 16-31: {2,3}
    const int rt    = wave >> 1;           // row tile 0..3
    const int ch    = wave & 1;            // col half 0..1
    const int rowA  = rt * 16 + l15;
    const int col0  = (ch * 2) * 16 + l15;         // first col tile's column
    const int col1  = (ch * 2 + 1) * 16 + l15;     // second col tile's column

    v8f acc0, acc1;
    {
        const float bv0 = bias_lds[col0];
        const float bv1 = bias_lds[col1];
        #pragma unroll
        for (int r = 0; r < 8; ++r) { acc0[r] = bv0; acc1[r] = bv1; }
    }

    #pragma unroll 4
    for (int kk = 0; kk < 32; ++kk) {
        const int kb = kk * 4 + khalf;
        v2f afrag;
        afrag.x = h_lds[rowA * LSTR + kb];
        afrag.y = h_lds[rowA * LSTR + kb + 1];
        v2f bfrag0, bfrag1;
        bfrag0.x = w_lds[col0 * LSTR + kb];
        bfrag0.y = w_lds[col0 * LSTR + kb + 1];
        bfrag1.x = w_lds[col1 * LSTR + kb];
        bfrag1.y = w_lds[col1 * LSTR + kb + 1];
        acc0 = __builtin_amdgcn_wmma_f32_16x16x4_f32(
            false, afrag, false, bfrag0, (short)0, acc0, false, false);
        acc1 = __builtin_amdgcn_wmma_f32_16x16x4_f32(
            false, afrag, false, bfrag1, (short)0, acc1, false, false);
    }

    // ---- store: D layout: VGPR r -> M = r + 8*(lane>=16), N = lane&15 ----
    // Only the last block can have out-of-range rows; keep the common path branch-free.
    const int mbase = blockIdx.x * ROWS + rt * 16 + (lane >> 4) * 8;
    float* __restrict__ op = out + (size_t)mbase * OUTC;
    const bool full = (blockIdx.x + 1) * ROWS <= NN * BB;
    if (full) {
        #pragma unroll
        for (int r = 0; r < 8; ++r) {
            op[(size_t)r * OUTC + col0] = acc0[r];
            op[(size_t)r * OUTC + col1] = acc1[r];
        }
    } else {
        #pragma unroll
        for (int r = 0; r < 8; ++r) {
            if (mbase + r < NN * BB) {
                op[(size_t)r * OUTC + col0] = acc0[r];
                op[(size_t)r * OUTC + col1] = acc1[r];
            }
        }
    }
}

extern "C" void kernel_launch(void* const* d_in, const int* in_sizes, int n_in,
                              void* d_out, int out_size, void* d_ws, size_t ws_size,
                              hipStream_t stream) {
    const float* x    = (const float*)d_in[0];  // (N, B, F)
    const float* dm   = (const float*)d_in[1];  // (N, NAS)
    const int*   idx  = (const int*)  d_in[2];  // (N, NAS)
    const float* W    = (const float*)d_in[3];  // (OUT, 2F)
    const float* bias = (const float*)d_in[4];  // (OUT,)
    float* out = (float*)d_out;                 // (N, B, OUT)

    const int grid = (NN * BB + ROWS - 1) / ROWS;  // 313
    pgnn_fused_kernel<<<grid, THREADS, 0, stream>>>(x, dm, idx, W, bias, out);
}